// lveg_7395933684364
// MI455X (gfx1250) — compile-verified
//
#include <hip/hip_runtime.h>
#include <stdint.h>

#define BB 8
#define TT 128
#define LL 32
#define DD 2

#define F_LOG_2PI   1.8378770664093453f
#define F_LOG2E     1.4426950408889634f
#define F_LN2       0.6931471805599453f

typedef float v4f __attribute__((ext_vector_type(4)));

__device__ __forceinline__ float fexp(float x) {            // e^x
    return __builtin_amdgcn_exp2f(x * F_LOG2E);
}
__device__ __forceinline__ float flog(float x) {            // ln x
    return __builtin_amdgcn_logf(x) * F_LN2;
}
__device__ __forceinline__ float frcp(float x) {
    return __builtin_amdgcn_rcpf(x);
}
__device__ __forceinline__ unsigned lds_off(const void* p) {
    // generic LDS pointer: low 32 bits are the wave-relative LDS byte address
    return (unsigned)(uintptr_t)p;
}

__global__ __launch_bounds__(1024) void lveg_kernel(
    const int*   __restrict__ input_ids,      // (B,T)
    const float* __restrict__ mask,           // (B,T)
    const float* __restrict__ trans_weight,   // (L,L)
    const float* __restrict__ trans_p_mu,     // (L,L,D)
    const float* __restrict__ trans_p_var,    // (L,L,D)
    const float* __restrict__ trans_c_mu,     // (L,L,D)
    const float* __restrict__ s_weight_table, // (V,1)
    const float* __restrict__ s_mu_table,     // (V,D)
    const float* __restrict__ s_var_table,    // (V,D)
    float*       __restrict__ out)            // (B,T,L,L,L)
{
    __shared__ __align__(16) float sPMU[LL * LL * DD];   // trans_p_mu         (8 KB, async)
    __shared__ __align__(16) float sEP2[LL * LL * DD];   // exp(2*trans_p_var) (8 KB)
    __shared__ __align__(16) float sTW [LL * LL];        // trans_weight       (4 KB, async)
    __shared__ v4f   sPACK[LL * LL];                     // {cs, mu0, mu1, w0} (16 KB)
    __shared__ float sA1  [LL * LL];                     // w1                 (4 KB)

    const int tid = threadIdx.x;           // 0..1023  == i*32+j
    const int blk = blockIdx.x;            // 0..B*T-1
    const int t   = blk % TT;

    // ---- Phase 1a: async DMA of k-invariant tables into LDS (ASYNCcnt path) ----
    if (tid < 512) {
        unsigned           l0 = lds_off(&sPMU[tid * 4]);
        unsigned long long g0 = (unsigned long long)(uintptr_t)(trans_p_mu + tid * 4);
        asm volatile("global_load_async_to_lds_b128 %0, %1, off"
                     :: "v"(l0), "v"(g0) : "memory");
    }
    if (tid < 256) {
        unsigned           l1 = lds_off(&sTW[tid * 4]);
        unsigned long long g1 = (unsigned long long)(uintptr_t)(trans_weight + tid * 4);
        asm volatile("global_load_async_to_lds_b128 %0, %1, off"
                     :: "v"(l1), "v"(g1) : "memory");
    }

    // ---- Phase 1b (overlapped): exp(2*trans_p_var) -> LDS ----
    {
        float pv0 = trans_p_var[tid * 2 + 0];
        float pv1 = trans_p_var[tid * 2 + 1];
        sEP2[tid * 2 + 0] = fexp(2.0f * pv0);
        sEP2[tid * 2 + 1] = fexp(2.0f * pv1);
    }

    // ---- Phase 2 (overlapped): first Gaussian product per (i,j) ----
    {
        const int tok = input_ids[blk];
        const float smu0  = s_mu_table [tok * 2 + 0];
        const float smu1  = s_mu_table [tok * 2 + 1];
        const float svar0 = s_var_table[tok * 2 + 0];
        const float svar1 = s_var_table[tok * 2 + 1];
        const float swt   = s_weight_table[tok];
        const float v1_0  = fexp(2.0f * svar0);
        const float v1_1  = fexp(2.0f * svar1);

        // faithful to the source bug: n2_var == trans_c_mu
        const float cmu0 = trans_c_mu[tid * 2 + 0];
        const float cmu1 = trans_c_mu[tid * 2 + 1];

        // d = 0
        float v2   = fexp(2.0f * cmu0);
        float vadd = v1_0 + v2;
        float lv   = flog(vadd);
        float rv   = frcp(vadd);
        float df   = smu0 - cmu0;
        float sc0  = F_LOG_2PI + lv + df * df * rv;
        float csmu0 = (smu0 * v2 + cmu0 * v1_0) * rv;
        float w0    = fexp(2.0f * (svar0 + cmu0 - 0.5f * lv));
        // d = 1
        v2   = fexp(2.0f * cmu1);
        vadd = v1_1 + v2;
        lv   = flog(vadd);
        rv   = frcp(vadd);
        df   = smu1 - cmu1;
        float sc1   = F_LOG_2PI + lv + df * df * rv;
        float csmu1 = (smu1 * v2 + cmu1 * v1_1) * rv;
        float w1    = fexp(2.0f * (svar1 + cmu1 - 0.5f * lv));

        v4f pk;
        pk.x = -0.5f * (sc0 + sc1) + swt;   // cs_scale (+ s_weight)
        pk.y = csmu0;
        pk.z = csmu1;
        pk.w = w0;
        sPACK[tid] = pk;
        sA1 [tid]  = w1;
    }

    asm volatile("s_wait_asynccnt 0" ::: "memory");
    __syncthreads();

    // ---- Phase 3: parent expansion; 8 lanes per (i,j), 4 k per lane, b128 NT stores ----
    const float mk = mask[blk];
    const int   pr = tid >> 3;             // local (i,j)-pair slot: 0..127
    const int   k0 = (tid & 7) << 2;       // k base: 0,4,...,28
    float* const outBase = out + (size_t)blk * (LL * LL * LL);

    if (t == TT - 1) {
        #pragma unroll
        for (int p = 0; p < 8; ++p) {
            const int g = pr + p * 128;    // global (i,j) index
            const float v = sPACK[g].x * mk;
            v4f val; val.x = v; val.y = v; val.z = v; val.w = v;
            __builtin_nontemporal_store(val, (v4f*)(outBase + g * 32 + k0));
        }
    } else {
        const float hm   = 0.5f * mk;
        const float cln2 = -0.5f * F_LN2 * mk;          // folds log2->ln and the -1/2
        #pragma unroll 2
        for (int p = 0; p < 8; ++p) {
            const int g  = pr + p * 128;
            const v4f pk = sPACK[g];                    // broadcast (8 lanes/addr)
            const float a1 = sA1[g];
            const float cs = pk.x, m0 = pk.y, m1 = pk.z, a0 = pk.w;
            const float base = mk * (cs - F_LOG_2PI);   // folds 2*LOG_2PI*(-1/2) + mask

            const v4f ep_lo = *(const v4f*)&sEP2[g * 64 + k0 * 2];
            const v4f ep_hi = *(const v4f*)&sEP2[g * 64 + k0 * 2 + 4];
            const v4f pm_lo = *(const v4f*)&sPMU[g * 64 + k0 * 2];
            const v4f pm_hi = *(const v4f*)&sPMU[g * 64 + k0 * 2 + 4];
            const v4f tw4   = *(const v4f*)&sTW [g * 32 + k0];

            const float b0e[4] = {ep_lo.x, ep_lo.z, ep_hi.x, ep_hi.z};
            const float b1e[4] = {ep_lo.y, ep_lo.w, ep_hi.y, ep_hi.w};
            const float p0e[4] = {pm_lo.x, pm_lo.z, pm_hi.x, pm_hi.z};
            const float p1e[4] = {pm_lo.y, pm_lo.w, pm_hi.y, pm_hi.w};
            const float twe[4] = {tw4.x, tw4.y, tw4.z, tw4.w};

            v4f val;
            #pragma unroll
            for (int e = 0; e < 4; ++e) {
                const float va0  = a0 + b0e[e];
                const float va1  = a1 + b1e[e];
                const float prod = va0 * va1;
                const float lg   = __builtin_amdgcn_logf(prod);  // log2: one v_log
                const float r    = __builtin_amdgcn_rcpf(prod);  // one v_rcp -> both invs
                const float d0   = m0 - p0e[e];
                const float d1   = m1 - p1e[e];
                const float q    = d0 * d0 * (r * va1) + d1 * d1 * (r * va0);
                float acc = fmaf(mk, twe[e], base);
                acc       = fmaf(-hm, q, acc);
                val[e]    = fmaf(cln2, lg, acc);
            }
            __builtin_nontemporal_store(val, (v4f*)(outBase + g * 32 + k0));
        }
    }
}

extern "C" void kernel_launch(void* const* d_in, const int* in_sizes, int n_in,
                              void* d_out, int out_size, void* d_ws, size_t ws_size,
                              hipStream_t stream) {
    (void)in_sizes; (void)n_in; (void)out_size; (void)d_ws; (void)ws_size;
    const int*   input_ids      = (const int*)  d_in[0];
    const float* mask           = (const float*)d_in[1];
    const float* trans_weight   = (const float*)d_in[2];
    const float* trans_p_mu     = (const float*)d_in[3];
    const float* trans_p_var    = (const float*)d_in[4];
    const float* trans_c_mu     = (const float*)d_in[5];
    // d_in[6] (trans_c_var) intentionally unused: reference has the c_mu-as-var bug
    const float* s_weight_table = (const float*)d_in[7];
    const float* s_mu_table     = (const float*)d_in[8];
    const float* s_var_table    = (const float*)d_in[9];
    float* out = (float*)d_out;

    dim3 grid(BB * TT);
    dim3 block(1024);
    lveg_kernel<<<grid, block, 0, stream>>>(
        input_ids, mask, trans_weight, trans_p_mu, trans_p_var, trans_c_mu,
        s_weight_table, s_mu_table, s_var_table, out);
}